// SparseAttention_89472758710437
// MI455X (gfx1250) — compile-verified
//
#include <hip/hip_runtime.h>
#include <hip/hip_bf16.h>

typedef __attribute__((ext_vector_type(2))) float v2f;
typedef __attribute__((ext_vector_type(8))) float v8f;

#define Ndim 4096
#define Edim 1024
#define Hh   16
#define Dd   64
#define Kk   64

__device__ __forceinline__ v8f wmma_f32(v2f a, v2f b, v8f c) {
    // D = A(16x4) * B(4x16) + C, fp32 everywhere
    return __builtin_amdgcn_wmma_f32_16x16x4_f32(
        false, a, false, b, (short)0, c, false, false);
}

__device__ __forceinline__ v2f load2(const float* p) {
    return *(const v2f*)p;
}

// monotone map: fp32 -> u32 so that descending float order == descending uint order
__device__ __forceinline__ unsigned fmap(float x) {
    unsigned u = __float_as_uint(x);
    return (u & 0x80000000u) ? ~u : (u | 0x80000000u);
}

// ---------------------------------------------------------------------------
// Kernel 1: projections.  Q = x @ Wq^T + bq  -> stored [H][N][64]
//                         K = x @ Wk^T + bk  -> stored [H][N][64] (row-major,
//                         which is exactly the WMMA B-fragment-friendly layout
//                         for q·k^T since lane pairs read consecutive d's)
// One wave computes a 32(M) x 64(E) tile; K-loop step 4 via WMMA f32 16x16x4.
// ---------------------------------------------------------------------------
__global__ __launch_bounds__(32) void proj_kernel(
    const float* __restrict__ x,
    const float* __restrict__ Wq, const float* __restrict__ bq,
    const float* __restrict__ Wk, const float* __restrict__ bk,
    float* __restrict__ Qws, float* __restrict__ Kws)
{
    const int lane = threadIdx.x;
    const int r  = lane & 15;      // row/col within 16
    const int hs = lane >> 4;      // which K-half this lane holds
    const int m0 = blockIdx.x * 32;
    const int e0 = blockIdx.y * 64;
    const bool isK = (blockIdx.z != 0);

    const float* W    = isK ? Wk : Wq;
    const float* bias = isK ? bk : bq;
    float*       dst  = isK ? Kws : Qws;

    v8f acc[2][4];
    #pragma unroll
    for (int i = 0; i < 2; ++i)
        #pragma unroll
        for (int t = 0; t < 4; ++t)
            acc[i][t] = (v8f){0.f,0.f,0.f,0.f,0.f,0.f,0.f,0.f};

    // A-frag: lane holds x[m0+r][c0+2*hs .. +1];  B-frag: lane holds W[e0+16t+r][c0+2*hs .. +1]
    for (int c0 = 0; c0 < Edim; c0 += 4) {
        v2f a0 = load2(x + (size_t)(m0 + r)      * Edim + c0 + 2 * hs);
        v2f a1 = load2(x + (size_t)(m0 + 16 + r) * Edim + c0 + 2 * hs);
        #pragma unroll
        for (int t = 0; t < 4; ++t) {
            v2f b = load2(W + (size_t)(e0 + 16 * t + r) * Edim + c0 + 2 * hs);
            acc[0][t] = wmma_f32(a0, b, acc[0][t]);
            acc[1][t] = wmma_f32(a1, b, acc[1][t]);
        }
    }

    // C/D layout: lane L, VGPR v holds C[M = v + 8*(L>>4)][Ncol = L&15]
    #pragma unroll
    for (int t = 0; t < 4; ++t) {
        const int e  = e0 + 16 * t + r;
        const float bv = bias[e];
        const int h = e >> 6;
        const int d = e & 63;
        #pragma unroll
        for (int i = 0; i < 2; ++i) {
            #pragma unroll
            for (int v = 0; v < 8; ++v) {
                const int row = m0 + 16 * i + v + 8 * hs;
                dst[((size_t)h * Ndim + row) * Dd + d] = acc[i][t][v] + bv; // [H][N][64]
            }
        }
    }
}

// ---------------------------------------------------------------------------
// Kernel 2: scores (WMMA) + exact top-64 radix select + dense write.
// Grid: (N/16, H). Block: 256 threads = 8 waves. Each wave computes a
// 16x512 strip of q·k^T*scale into LDS; then each wave selects top-64 for
// 2 of the 16 rows and writes the full dense output row exactly once.
// ---------------------------------------------------------------------------
__global__ __launch_bounds__(256) void attn_kernel(
    const float* __restrict__ Qws,
    const float* __restrict__ Kws,
    float* __restrict__ out)
{
    extern __shared__ unsigned char smem[];
    float*    sc      = (float*)smem;                                   // [16][4096]
    unsigned* histAll = (unsigned*)(smem + 16 * 4096 * sizeof(float));  // [8][256]
    unsigned* ctr     = histAll + 8 * 256;                              // [16]

    const int tid  = threadIdx.x;
    const int lane = tid & 31;
    const int wave = tid >> 5;
    const int r  = lane & 15;
    const int hs = lane >> 4;
    const int n0 = blockIdx.x * 16;
    const int h  = blockIdx.y;

    const float* qp = Qws + ((size_t)h * Ndim + n0) * Dd;
    const float* kp = Kws + (size_t)h * Ndim * Dd;

    // ---- Phase 1: scores strip via WMMA (uniform EXEC) ----
    v2f afr[16];
    #pragma unroll
    for (int s = 0; s < 16; ++s)
        afr[s] = load2(qp + (size_t)r * Dd + 4 * s + 2 * hs);

    for (int t = 0; t < 32; ++t) {
        const int key0 = wave * 512 + t * 16;
        if (t < 31)  // prefetch next key tile's row (global_prefetch_b8)
            __builtin_prefetch(kp + (size_t)(key0 + 16 + r) * Dd + 2 * hs, 0, 1);

        // B-frag: lane holds k[key0+r][4s+2hs .. +1]  (coalesced b64 like A)
        const float* brow = kp + (size_t)(key0 + r) * Dd + 2 * hs;

        v8f acc = (v8f){0.f,0.f,0.f,0.f,0.f,0.f,0.f,0.f};
        #pragma unroll
        for (int s = 0; s < 16; ++s) {
            v2f b = load2(brow + 4 * s);
            acc = wmma_f32(afr[s], b, acc);
        }
        #pragma unroll
        for (int v = 0; v < 8; ++v)
            sc[(v + 8 * hs) * Ndim + key0 + r] = acc[v] * 0.125f;
    }
    __syncthreads();

    // ---- Phase 2/3: per-row exact top-64 (radix select) + dense write ----
    unsigned* hist = histAll + wave * 256;

    for (int rr = 0; rr < 2; ++rr) {
        const int row = wave * 2 + rr;
        const float* rowp = sc + (size_t)row * Ndim;

        unsigned prefix = 0, himask = 0;
        int kneed = Kk;

        for (int pass = 0; pass < 4; ++pass) {
            const int shift = 24 - 8 * pass;
            for (int i = lane; i < 256; i += 32) hist[i] = 0;
            __syncthreads();
            for (int j = 0; j < 128; ++j) {
                unsigned u = fmap(rowp[lane + 32 * j]);
                if ((u & himask) == prefix)
                    atomicAdd(&hist[(u >> shift) & 255u], 1u);
            }
            __syncthreads();
            // uniform scan from the top digit down
            int cum = 0, chosen = 0;
            for (int b = 255; b >= 0; --b) {
                int c = (int)hist[b];
                if (cum + c >= kneed) { chosen = b; break; }
                cum += c;
            }
            kneed -= cum;
            prefix |= (unsigned)chosen << shift;
            himask |= 0xFFu << shift;
            __syncthreads();
        }

        if (lane == 0) ctr[row] = 0;
        __syncthreads();

        const unsigned uth = prefix;       // exact bit pattern of the 64th largest
        float* obase = out + ((size_t)h * Ndim + (n0 + row)) * Ndim;
        for (int j = 0; j < 128; ++j) {
            const int idx = lane + 32 * j;
            const float f = rowp[idx];
            const unsigned u = fmap(f);
            float val = 0.0f;
            if (u > uth) {
                val = f;
            } else if (u == uth) {
                unsigned slot = atomicAdd(&ctr[row], 1u);
                if (slot < (unsigned)kneed) val = f;   // tie handling: take first 'kneed'
            }
            obase[idx] = val;   // every output element written exactly once
        }
        __syncthreads();
    }
}

// ---------------------------------------------------------------------------
extern "C" void kernel_launch(void* const* d_in, const int* in_sizes, int n_in,
                              void* d_out, int out_size, void* d_ws, size_t ws_size,
                              hipStream_t stream) {
    (void)in_sizes; (void)n_in; (void)out_size; (void)ws_size;
    const float* x  = (const float*)d_in[0];
    const float* Wq = (const float*)d_in[1];
    const float* bq = (const float*)d_in[2];
    const float* Wk = (const float*)d_in[3];
    const float* bk = (const float*)d_in[4];
    float* out = (float*)d_out;

    float* Qws = (float*)d_ws;                       // [H][N][64]  = 16 MB
    float* Kws = Qws + (size_t)Ndim * Edim;          // [H][N][64]  = 16 MB

    // Projections: grid over (M tiles of 32, E tiles of 64, {Q,K})
    dim3 pg(Ndim / 32, Edim / 64, 2);
    proj_kernel<<<pg, dim3(32), 0, stream>>>(x, Wq, bq, Wk, bk, Qws, Kws);

    // Scores + top-k + dense write
    size_t shmem = 16 * 4096 * sizeof(float)   // scores strip (256 KB)
                 + 8 * 256 * sizeof(unsigned)  // per-wave histograms
                 + 16 * sizeof(unsigned);      // per-row tie counters
    dim3 ag(Ndim / 16, Hh);
    attn_kernel<<<ag, dim3(256), shmem, stream>>>(Qws, Kws, out);
}